// Eq2to2_58815282152312
// MI455X (gfx1250) — compile-verified
//
#include <hip/hip_runtime.h>

typedef __attribute__((ext_vector_type(2))) float v2f;
typedef __attribute__((ext_vector_type(8))) float v8f;

#define Bn   8
#define Nn   128
#define CIN  64
#define COUT 64
#define NBAS 15
#define AVG  49.0f
#define NEG_SLOPE 0.01f

// ---- workspace layout (in floats) ----
#define OFF_C      0                       // 15 * 64 * 64 = 61440
#define OFF_DG     (OFF_C + NBAS*CIN*COUT) // 8*128*64 = 65536
#define OFF_RS     (OFF_DG + Bn*Nn*CIN)
#define OFF_CS     (OFF_RS + Bn*Nn*CIN)
#define OFF_TR     (OFF_CS + Bn*Nn*CIN)    // 8*64
#define OFF_AS     (OFF_TR + Bn*CIN)
#define OFF_RV     (OFF_AS + Bn*CIN)       // 8*128*64
#define OFF_CV     (OFF_RV + Bn*Nn*COUT)
#define OFF_DV     (OFF_CV + Bn*Nn*COUT)
#define OFF_SC     (OFF_DV + Bn*Nn*COUT)   // 8*64
#define OFF_DSC    (OFF_SC + Bn*COUT)

// ---------------------------------------------------------------------------
// Kernel 0: reconstruct the 15 coefficient matrices C_b[d,s], folding in the
// 1/49 and 1/49^2 normalizations of the aggregation ops.
// basis order: 0 diag*eye, 1 A_ii bc, 2 A_jj bc, 3 identity, 4 transpose,
// 5 rowsum*eye, 6 colsum*eye, 7 trace*eye, 8 allsum*eye,
// 9 rowsum_i, 10 rowsum_j, 11 colsum_i, 12 colsum_j, 13 trace bc, 14 allsum bc
// ---------------------------------------------------------------------------
__global__ void k_coefs(const float* __restrict__ c00, const float* __restrict__ c01,
                        const float* __restrict__ c10, const float* __restrict__ c11,
                        float* __restrict__ C) {
  const int d = blockIdx.x;   // C_IN
  const int s = threadIdx.x;  // C_OUT
  const float a10 = c10[d*COUT + s];
  const float a11 = c11[d*COUT + s];
  const float inv = 1.0f/AVG, inv2 = inv*inv;
  const float norm[NBAS] = {1.f,1.f,1.f,1.f,1.f,
                            inv,inv,inv,inv2,
                            inv,inv,inv,inv,
                            inv,inv2};
#pragma unroll
  for (int b = 0; b < NBAS; ++b) {
    float v = c00[d*NBAS + b]*a10 + c01[s*NBAS + b]*a11;
    C[b*CIN*COUT + d*COUT + s] = v * norm[b];
  }
}

// ---------------------------------------------------------------------------
// Kernel 1: per-(n, index p, channel d) reductions: diag, raw row/col sums.
// grid (Nn, Bn), block 64 (d)
// ---------------------------------------------------------------------------
__global__ void k_reduce(const float* __restrict__ in,
                         float* __restrict__ dg, float* __restrict__ rs,
                         float* __restrict__ cs) {
  const int p = blockIdx.x, n = blockIdx.y, d = threadIdx.x;
  const float* base = in + (size_t)n*Nn*Nn*CIN;
  float r = 0.f, c = 0.f;
  for (int q = 0; q < Nn; ++q) {
    r += base[(p*Nn + q)*CIN + d];   // row sum over j
    c += base[(q*Nn + p)*CIN + d];   // col sum over i
  }
  const int idx = (n*Nn + p)*CIN + d;
  rs[idx] = r;
  cs[idx] = c;
  dg[idx] = base[(p*Nn + p)*CIN + d];
}

// Kernel 2: trace / allsum (raw). grid Bn, block 64
__global__ void k_tr(const float* __restrict__ dg, const float* __restrict__ rs,
                     float* __restrict__ tr, float* __restrict__ as_) {
  const int n = blockIdx.x, d = threadIdx.x;
  float t = 0.f, a = 0.f;
  for (int i = 0; i < Nn; ++i) {
    t += dg[(n*Nn + i)*CIN + d];
    a += rs[(n*Nn + i)*CIN + d];
  }
  tr[n*CIN + d] = t;
  as_[n*CIN + d] = a;
}

// ---------------------------------------------------------------------------
// Kernel 3a: broadcast vectors rowvec/colvec/diagvec [B,N,COUT].
// grid Bn*Nn, block 64 (s)
// ---------------------------------------------------------------------------
__global__ void k_vec(const float* __restrict__ dg, const float* __restrict__ rs,
                      const float* __restrict__ cs, const float* __restrict__ C,
                      float* __restrict__ rowv, float* __restrict__ colv,
                      float* __restrict__ diagv) {
  const int ni = blockIdx.x;   // n*Nn + i
  const int s  = threadIdx.x;
  __shared__ float sdg[CIN], srs[CIN], scs[CIN];
  sdg[s] = dg[ni*CIN + s];
  srs[s] = rs[ni*CIN + s];
  scs[s] = cs[ni*CIN + s];
  __syncthreads();
  const float* C1  = C + 0*CIN*COUT;
  const float* C2  = C + 1*CIN*COUT;
  const float* C3  = C + 2*CIN*COUT;
  const float* C6  = C + 5*CIN*COUT;
  const float* C7  = C + 6*CIN*COUT;
  const float* C10 = C + 9*CIN*COUT;
  const float* C11 = C + 10*CIN*COUT;
  const float* C12 = C + 11*CIN*COUT;
  const float* C13 = C + 12*CIN*COUT;
  float rv = 0.f, cv = 0.f, dv = 0.f;
  for (int d = 0; d < CIN; ++d) {
    const float dd = sdg[d], rr = srs[d], cc = scs[d];
    const int o = d*COUT + s;
    rv += dd*C2[o] + rr*C10[o] + cc*C12[o];
    cv += dd*C3[o] + rr*C11[o] + cc*C13[o];
    dv += dd*C1[o] + rr*C6[o]  + cc*C7[o];
  }
  rowv[ni*COUT + s] = rv;
  colv[ni*COUT + s] = cv;
  diagv[ni*COUT + s] = dv;
}

// Kernel 3b: scalar / diag-scalar per (n,s). grid Bn, block 64
__global__ void k_scal(const float* __restrict__ tr, const float* __restrict__ as_,
                       const float* __restrict__ C, const float* __restrict__ bias,
                       const float* __restrict__ dbias,
                       float* __restrict__ sc, float* __restrict__ dsc) {
  const int n = blockIdx.x, s = threadIdx.x;
  const float* C8  = C + 7*CIN*COUT;
  const float* C9  = C + 8*CIN*COUT;
  const float* C14 = C + 13*CIN*COUT;
  const float* C15 = C + 14*CIN*COUT;
  float a = bias[s], b = dbias[s];
  for (int d = 0; d < CIN; ++d) {
    const float t = tr[n*CIN + d], u = as_[n*CIN + d];
    const int o = d*COUT + s;
    a += t*C14[o] + u*C15[o];
    b += t*C8[o]  + u*C9[o];
  }
  sc[n*COUT + s]  = a;
  dsc[n*COUT + s] = b;
}

// ---------------------------------------------------------------------------
// Main kernel: per-pixel dual GEMM via V_WMMA_F32_16X16X4_F32 + epilogue.
// grid (Nn/16 j-tiles, Nn/16 i-tiles, Bn), block 128 (4 waves; wave w -> s0=16w)
// A layout (16x4 f32): lane L holds row M=L&15; VGPR0 = K=2*(L>>4),
//                      VGPR1 = K=2*(L>>4)+1  -> float2 global load.
// B layout (4x16 f32): lane L holds col N=L&15; VGPR0 = row K=2*(L>>4),
//                      VGPR1 = row K=2*(L>>4)+1.
// C/D layout (16x16 f32): lane L, reg r -> (M = r + 8*(L>>4), N = L&15).
// ---------------------------------------------------------------------------
__global__ __launch_bounds__(128, 1)
void k_main(const float* __restrict__ in, const float* __restrict__ mask,
            const float* __restrict__ C,
            const float* __restrict__ rowv, const float* __restrict__ colv,
            const float* __restrict__ diagv, const float* __restrict__ sc,
            const float* __restrict__ dsc, float* __restrict__ out) {
  const int lane = threadIdx.x & 31;
  const int wave = threadIdx.x >> 5;
  const int lo = lane & 15;
  const int hi = lane >> 4;
  const int s0 = wave * 16;
  const int scol = s0 + lo;
  const int n  = blockIdx.z;
  const int it = blockIdx.y;
  const int j0 = blockIdx.x * 16;

  const float* C4 = C + 3*CIN*COUT;  // identity term
  const float* C5 = C + 4*CIN*COUT;  // transpose term

  // Hoist all B fragments for this wave's s-block (constant over i-loop).
  v2f b4[16], b5[16];
#pragma unroll
  for (int kk = 0; kk < 16; ++kk) {
    const int d0 = 4*kk + 2*hi;
    v2f t4 = { C4[d0*COUT + scol], C4[(d0+1)*COUT + scol] };
    v2f t5 = { C5[d0*COUT + scol], C5[(d0+1)*COUT + scol] };
    b4[kk] = t4;
    b5[kk] = t5;
  }

  const int jr = j0 + lo;  // A-fragment row: pixel j index

  for (int ii = 0; ii < 16; ++ii) {
    const int i = it*16 + ii;
    const float* aXbase = in + ((size_t)(n*Nn + i )*Nn + jr)*CIN + 2*hi;
    const float* aTbase = in + ((size_t)(n*Nn + jr)*Nn + i )*CIN + 2*hi;

    v2f aX[16], aT[16];
#pragma unroll
    for (int kk = 0; kk < 16; ++kk) {
      aX[kk] = *(const v2f*)(aXbase + 4*kk);
      aT[kk] = *(const v2f*)(aTbase + 4*kk);
    }

    v8f acc = {};
#pragma unroll
    for (int kk = 0; kk < 16; ++kk) {
      acc = __builtin_amdgcn_wmma_f32_16x16x4_f32(
          false, aX[kk], false, b4[kk], (short)0, acc, false, false);
      acc = __builtin_amdgcn_wmma_f32_16x16x4_f32(
          false, aT[kk], false, b5[kk], (short)0, acc, false, false);
    }

    // epilogue: broadcast terms + diagonal + bias (in sc/dsc) + lrelu + mask
    const float rvi = rowv [(n*Nn + i)*COUT + scol];
    const float dvi = diagv[(n*Nn + i)*COUT + scol] + dsc[n*COUT + scol];
    const float sci = sc[n*COUT + scol];
#pragma unroll
    for (int r = 0; r < 8; ++r) {
      const int j = j0 + r + 8*hi;          // D-matrix row M = r + 8*hi
      float v = acc[r] + rvi + colv[(n*Nn + j)*COUT + scol] + sci;
      if (i == j) v += dvi;
      v = (v > 0.f) ? v : v*NEG_SLOPE;
      v *= mask[(n*Nn + i)*Nn + j];
      out[((size_t)(n*Nn + i)*Nn + j)*COUT + scol] = v;
    }
  }
}

// ---------------------------------------------------------------------------
extern "C" void kernel_launch(void* const* d_in, const int* in_sizes, int n_in,
                              void* d_out, int out_size, void* d_ws, size_t ws_size,
                              hipStream_t stream) {
  (void)in_sizes; (void)n_in; (void)out_size; (void)ws_size;
  const float* inputs = (const float*)d_in[0];
  const float* mask   = (const float*)d_in[1];
  // d_in[2] = nobj (unused by reference math; AVG_NOBJ is a constant)
  const float* c00    = (const float*)d_in[3];
  const float* c01    = (const float*)d_in[4];
  const float* c10    = (const float*)d_in[5];
  const float* c11    = (const float*)d_in[6];
  const float* bias   = (const float*)d_in[7];
  const float* dbias  = (const float*)d_in[8];
  float* out = (float*)d_out;
  float* ws  = (float*)d_ws;

  float* C    = ws + OFF_C;
  float* dg   = ws + OFF_DG;
  float* rs   = ws + OFF_RS;
  float* cs   = ws + OFF_CS;
  float* tr   = ws + OFF_TR;
  float* as_  = ws + OFF_AS;
  float* rowv = ws + OFF_RV;
  float* colv = ws + OFF_CV;
  float* dgv  = ws + OFF_DV;
  float* sc   = ws + OFF_SC;
  float* dsc  = ws + OFF_DSC;

  k_coefs <<<CIN, COUT, 0, stream>>>(c00, c01, c10, c11, C);
  k_reduce<<<dim3(Nn, Bn), CIN, 0, stream>>>(inputs, dg, rs, cs);
  k_tr    <<<Bn, CIN, 0, stream>>>(dg, rs, tr, as_);
  k_vec   <<<Bn*Nn, COUT, 0, stream>>>(dg, rs, cs, C, rowv, colv, dgv);
  k_scal  <<<Bn, COUT, 0, stream>>>(tr, as_, C, bias, dbias, sc, dsc);
  k_main  <<<dim3(Nn/16, Nn/16, Bn), 128, 0, stream>>>(
      inputs, mask, C, rowv, colv, dgv, sc, dsc, out);
}